// WaveNet_7438883357532
// MI455X (gfx1250) — compile-verified
//
#include <hip/hip_runtime.h>

typedef float v2f __attribute__((ext_vector_type(2)));
typedef float v8f __attribute__((ext_vector_type(8)));

#define WN_C 16
#define WN_T 131072
#define WN_HALO 1024                    // = max 2*dil, causal zero pad on the left
#define WN_THS (WN_T + WN_HALO)
#define WPK_PER_LAYER 2048              // floats: 6 conv mats + skip + res

// ---------------------------------------------------------------------------
// repack weights into per-lane WMMA A-fragments.
// layout: wpk[layer][mat][chunk][lane][j]  (mat: k*2+p for conv, 6=skip, 7=res)
// fragment element (lane l, comp j) = W[row M=l&15][K = 4*chunk + 2*(l>>4) + j]
// ---------------------------------------------------------------------------
__global__ __launch_bounds__(256) void wavenet_repack(
    const float* __restrict__ cw, const float* __restrict__ rw,
    const float* __restrict__ sw, float* __restrict__ wpk)
{
  int layer = blockIdx.x;
  const float* cwl = cw + (size_t)layer * 32 * 16 * 3;
  const float* rwl = rw + (size_t)layer * 256;
  const float* swl = sw + (size_t)layer * 256;
  float* o = wpk + (size_t)layer * WPK_PER_LAYER;
  for (int f = threadIdx.x; f < WPK_PER_LAYER; f += 256) {
    int j   = f & 1;
    int l   = (f >> 1) & 31;
    int M   = l & 15;
    int hpv = l >> 4;
    int rest = f >> 6;        // mat*4 + chunk
    int c   = rest & 3;
    int mat = rest >> 2;      // 0..5 conv (k*2+p), 6 skip, 7 res
    int K   = 4 * c + 2 * hpv + j;
    float val;
    if (mat < 6) {
      int k = mat >> 1, p = mat & 1;
      val = cwl[((p * 16 + M) * 16 + K) * 3 + k];
    } else if (mat == 6) {
      val = swl[M * 16 + K];
    } else {
      val = rwl[M * 16 + K];
    }
    o[f] = val;
  }
}

// ---------------------------------------------------------------------------
// init. Buffers pair-interleaved with halo: buf[b][pair][HALO + t][comp].
// ---------------------------------------------------------------------------
__global__ __launch_bounds__(256) void wavenet_init(
    const float* __restrict__ x, const float* __restrict__ in_w,
    const float* __restrict__ in_b, float* __restrict__ h0,
    float* __restrict__ h1, float* __restrict__ S)
{
  const unsigned bx = blockIdx.x;
  __builtin_assume(bx < (WN_THS + 255u) / 256u);
  int th = bx * 256 + threadIdx.x;            // position incl. halo
  int b  = blockIdx.y;
  if (th >= WN_THS) return;
  float* h0b = h0 + (size_t)b * 8 * WN_THS * 2;
  float* h1b = h1 + (size_t)b * 8 * WN_THS * 2;
  float* Sb  = S  + (size_t)b * 8 * WN_THS * 2;
  if (th < WN_HALO) {
    // zero left halo of both ping-pong h buffers (stays zero forever)
#pragma unroll
    for (int pr = 0; pr < 8; ++pr) {
      unsigned base = ((unsigned)pr * WN_THS + (unsigned)th) * 2u;
      *(v2f*)(h0b + base) = (v2f){0.0f, 0.0f};
      *(v2f*)(h1b + base) = (v2f){0.0f, 0.0f};
    }
    return;
  }
  float xv = x[(size_t)b * WN_T + (th - WN_HALO)];
#pragma unroll
  for (int pr = 0; pr < 8; ++pr) {
    unsigned base = ((unsigned)pr * WN_THS + (unsigned)th) * 2u;
    v2f hv = (v2f){ fmaf(in_w[2 * pr],     xv, in_b[2 * pr]),
                    fmaf(in_w[2 * pr + 1], xv, in_b[2 * pr + 1]) };
    *(v2f*)(h0b + base) = hv;
    *(v2f*)(Sb  + base) = (v2f){0.0f, 0.0f};
  }
}

// ---------------------------------------------------------------------------
// one WaveNet layer (DIL = compile-time dilation): per wave, one 16ch x 16t
// tile via f32 WMMA chains. All offsets are provably-bounded 32-bit values so
// tap/epilogue loads fold into saddr + immediate-offset form.
// ---------------------------------------------------------------------------
template <int DIL>
__global__ __launch_bounds__(256) void wavenet_layer(
    const float* __restrict__ h_in, float* __restrict__ h_out,
    float* __restrict__ S,
    const float* __restrict__ wl,   // repacked fragments for this layer
    const float* __restrict__ cb,   // [32]
    const float* __restrict__ rb,   // [16]
    const float* __restrict__ sb)   // [16]
{
  __shared__ float lds[8][16 * 20];

  const unsigned bx = blockIdx.x;
  __builtin_assume(bx < WN_T / 128);
  const int wave = threadIdx.x >> 5;
  const int lane = threadIdx.x & 31;
  const int hp   = lane >> 4;
  const int n    = lane & 15;
  const int b    = blockIdx.y;
  const unsigned tbase = (bx * 8 + wave) * 16;

  // uniform (scalar) per-batch bases
  const float* hb_in  = h_in  + (size_t)b * 8 * WN_THS * 2;
  float*       hb_out = h_out + (size_t)b * 8 * WN_THS * 2;
  float*       Sb     = S     + (size_t)b * 8 * WN_THS * 2;

  // ---- bias fragments (C/D layout) ----------------------------------------
  v8f accf = *(const v8f*)(cb + 8 * hp);         // tanh half   (out 0..15)
  v8f accg = *(const v8f*)(cb + 16 + 8 * hp);    // sigmoid half(out 16..31)
  v8f accs = *(const v8f*)(sb + 8 * hp);
  v8f accr = *(const v8f*)(rb + 8 * hp);

  // ---- weight A-fragments: coalesced b64 loads, no shuffles ----------------
  v2f Acv[3][2][4];                              // [tap][half][chunk]
#pragma unroll
  for (int k = 0; k < 3; ++k)
#pragma unroll
    for (int p = 0; p < 2; ++p)
#pragma unroll
      for (int c = 0; c < 4; ++c)
        Acv[k][p][c] = *(const v2f*)(wl + (((k * 2 + p) * 4 + c) * 32 + lane) * 2);
  v2f Ask[4], Are[4];
#pragma unroll
  for (int c = 0; c < 4; ++c) {
    Ask[c] = *(const v2f*)(wl + ((6 * 4 + c) * 32 + lane) * 2);
    Are[c] = *(const v2f*)(wl + ((7 * 4 + c) * 32 + lane) * 2);
  }

  // ---- 3 dilated tap tiles, B layout --------------------------------------
  // per-lane part: (hp*THS + HALO + tbase + n); per-(k,c) part is constant
  const unsigned vbase = (unsigned)hp * WN_THS + WN_HALO + tbase + (unsigned)n;
  v2f Btap[3][4];
#pragma unroll
  for (int k = 0; k < 3; ++k) {
#pragma unroll
    for (int c = 0; c < 4; ++c) {
      unsigned off = (vbase + (unsigned)(2 * c) * WN_THS
                            - (unsigned)((2 - k) * DIL)) * 2u;
      Btap[k][c] = *(const v2f*)(hb_in + off);
    }
  }

#pragma unroll
  for (int k = 0; k < 3; ++k)
#pragma unroll
    for (int c = 0; c < 4; ++c) {
      accf = __builtin_amdgcn_wmma_f32_16x16x4_f32(false, Acv[k][0][c], false,
                 Btap[k][c], (short)0, accf, false, false);
      accg = __builtin_amdgcn_wmma_f32_16x16x4_f32(false, Acv[k][1][c], false,
                 Btap[k][c], (short)0, accg, false, false);
    }

  // ---- gated = tanh * sigmoid (fast exp/rcp, clamped) ----------------------
  v8f g;
#pragma unroll
  for (int v = 0; v < 8; ++v) {
    float f  = fminf(fmaxf(accf[v], -10.0f), 10.0f);
    float e2 = __expf(2.0f * f);
    float th = (e2 - 1.0f) * __builtin_amdgcn_rcpf(e2 + 1.0f);
    float sg = __builtin_amdgcn_rcpf(1.0f + __expf(-accg[v]));
    g[v] = th * sg;
  }

  // ---- D-layout -> B-layout via conflict-free LDS --------------------------
  float* ld = lds[wave];
#pragma unroll
  for (int v = 0; v < 8; ++v)
    ld[(v + 8 * hp) * 20 + n] = g[v];
  __syncthreads();

  v2f Bg[4];
#pragma unroll
  for (int c = 0; c < 4; ++c) {
    int kk = 4 * c + 2 * hp;
    Bg[c] = (v2f){ ld[kk * 20 + n], ld[(kk + 1) * 20 + n] };
  }

  // ---- skip / residual 1x1 convs ------------------------------------------
#pragma unroll
  for (int c = 0; c < 4; ++c) {
    accs = __builtin_amdgcn_wmma_f32_16x16x4_f32(false, Ask[c], false, Bg[c],
                                                 (short)0, accs, false, false);
    accr = __builtin_amdgcn_wmma_f32_16x16x4_f32(false, Are[c], false, Bg[c],
                                                 (short)0, accr, false, false);
  }

  // ---- global updates: vectorized b64 RMW, shared voffset + imm offsets ----
  const unsigned ebase = ((unsigned)(4 * hp) * WN_THS + WN_HALO + tbase
                          + (unsigned)n) * 2u;
#pragma unroll
  for (int vp = 0; vp < 4; ++vp) {
    unsigned base = ebase + (unsigned)vp * (WN_THS * 2u);
    v2f sv = *(const v2f*)(Sb + base);
    sv.x += accs[2 * vp];
    sv.y += accs[2 * vp + 1];
    *(v2f*)(Sb + base) = sv;
    v2f hv = *(const v2f*)(hb_in + base);
    hv.x = fmaf(0.1f, accr[2 * vp],     hv.x);
    hv.y = fmaf(0.1f, accr[2 * vp + 1], hv.y);
    *(v2f*)(hb_out + base) = hv;
  }
}

// ---------------------------------------------------------------------------
// out[b][0][t] = out_w . S[b][:][t] + out_b   (S pair-interleaved, halo stride)
// ---------------------------------------------------------------------------
__global__ __launch_bounds__(256) void wavenet_out(
    const float* __restrict__ S, const float* __restrict__ out_w,
    const float* __restrict__ out_b, float* __restrict__ out)
{
  const unsigned bx = blockIdx.x;
  __builtin_assume(bx < WN_T / 256);
  int t = bx * 256 + threadIdx.x;
  int b = blockIdx.y;
  const float* Sb = S + (size_t)b * 8 * WN_THS * 2;
  float acc = out_b[0];
#pragma unroll
  for (int pr = 0; pr < 8; ++pr) {
    unsigned off = ((unsigned)pr * WN_THS + (unsigned)(WN_HALO + t)) * 2u;
    v2f sv = *(const v2f*)(Sb + off);
    acc = fmaf(out_w[2 * pr], sv.x, fmaf(out_w[2 * pr + 1], sv.y, acc));
  }
  out[(size_t)b * WN_T + t] = acc;
}

// ---------------------------------------------------------------------------
extern "C" void kernel_launch(void* const* d_in, const int* in_sizes, int n_in,
                              void* d_out, int out_size, void* d_ws, size_t ws_size,
                              hipStream_t stream) {
  const float* x    = (const float*)d_in[0];
  const float* in_w = (const float*)d_in[1];
  const float* in_b = (const float*)d_in[2];
  const float* cw   = (const float*)d_in[3];   // [10][32][16][3]
  const float* cb   = (const float*)d_in[4];   // [10][32]
  const float* rw   = (const float*)d_in[5];   // [10][16][16]
  const float* rb   = (const float*)d_in[6];   // [10][16]
  const float* sw   = (const float*)d_in[7];   // [10][16][16]
  const float* sb   = (const float*)d_in[8];   // [10][16]
  const float* ow   = (const float*)d_in[9];   // [1][16]
  const float* ob   = (const float*)d_in[10];  // [1]

  const int B = in_sizes[0] / WN_T;
  const int L = 10;
  const size_t NP = (size_t)B * WN_C * WN_THS;  // floats per padded buffer

  // workspace: H0 | H1 (ping-pong hidden state) | S | repacked weights
  float* H0   = (float*)d_ws;
  float* H1   = H0 + NP;
  float* Ssum = H0 + 2 * NP;
  float* Wpk  = H0 + 3 * NP;

  wavenet_repack<<<dim3(L), dim3(256), 0, stream>>>(cw, rw, sw, Wpk);
  wavenet_init<<<dim3((WN_THS + 255) / 256, B), dim3(256), 0, stream>>>(
      x, in_w, in_b, H0, H1, Ssum);

  float* hin = H0;
  float* hout = H1;
  dim3 lgrid(WN_T / 128, B), lblk(256);
#define WN_LAUNCH(D)                                                         \
    wavenet_layer<D><<<lgrid, lblk, 0, stream>>>(                            \
        hin, hout, Ssum, Wpk + (size_t)i * WPK_PER_LAYER,                    \
        cb + (size_t)i * 32, rb + (size_t)i * 16, sb + (size_t)i * 16)
  for (int i = 0; i < L; ++i) {
    switch (i) {
      case 0: WN_LAUNCH(1);   break;
      case 1: WN_LAUNCH(2);   break;
      case 2: WN_LAUNCH(4);   break;
      case 3: WN_LAUNCH(8);   break;
      case 4: WN_LAUNCH(16);  break;
      case 5: WN_LAUNCH(32);  break;
      case 6: WN_LAUNCH(64);  break;
      case 7: WN_LAUNCH(128); break;
      case 8: WN_LAUNCH(256); break;
      case 9: WN_LAUNCH(512); break;
    }
    float* tmp = hin; hin = hout; hout = tmp;
  }
#undef WN_LAUNCH

  wavenet_out<<<dim3(WN_T / 256, B), dim3(256), 0, stream>>>(
      Ssum, ow, ob, (float*)d_out);
}